// Linear_81544249082264
// MI455X (gfx1250) — compile-verified
//
#include <hip/hip_runtime.h>

// Sorted-COO SpMV:  out[r] = sum_{i in row r} values[i] * w[feat_ids[i]] + b[0]
// One wave32 per row; scalar (SGPR) binary search for the row segment;
// values/feat_ids streamed HBM -> LDS via gfx1250 async-to-LDS double buffer;
// w[] gathers hit L2 (4MB << 192MB); deterministic shuffle reduction.

#define TILE 128            // elements per staged tile per wave
#define WAVES_PER_BLOCK 8   // blockDim 256 / wave32

__device__ __forceinline__ long long lower_bound_i32(const int* __restrict__ a,
                                                     long long n, int key) {
    long long lo = 0, hi = n;
    while (lo < hi) {
        long long mid = (lo + hi) >> 1;
        if (a[mid] < key) lo = mid + 1; else hi = mid;
    }
    return lo;
}

__device__ __forceinline__ void async_g2lds_b32(const void* gaddr, const void* laddr) {
    unsigned long long g = (unsigned long long)(uintptr_t)gaddr;
    unsigned           l = (unsigned)(uintptr_t)laddr;  // LDS offset = addr[31:0]
    asm volatile("global_load_async_to_lds_b32 %0, %1, off"
                 :: "v"(l), "v"(g) : "memory");
}

__global__ __launch_bounds__(256) void fm_linear_spmv_rowwave(
    const float* __restrict__ values,
    const float* __restrict__ w,        // [F,1] -> flat [F]
    const float* __restrict__ b,        // [1]
    const int*   __restrict__ row_ids,  // sorted, len nnz
    const int*   __restrict__ feat_ids, // len nnz
    float*       __restrict__ out,      // [batch]
    long long nnz, int batch)
{
    __shared__ float s_vals[WAVES_PER_BLOCK][2][TILE];
    __shared__ int   s_feat[WAVES_PER_BLOCK][2][TILE];

    const int lane = threadIdx.x & 31;
    const int wave = threadIdx.x >> 5;
    const int gwave = (int)((blockIdx.x * (long long)blockDim.x + threadIdx.x) >> 5);
    // Wave-uniform row index -> SGPR, so the binary search scalarizes (s_load).
    const int row = __builtin_amdgcn_readfirstlane(gwave);
    if (row >= batch) return;

    const long long lo = lower_bound_i32(row_ids, nnz, row);
    const long long hi = lower_bound_i32(row_ids, nnz, row + 1);
    const long long n  = hi - lo;
    const long long nfull = n >> 7;   // full 128-elem tiles

    float a0 = 0.f, a1 = 0.f, a2 = 0.f, a3 = 0.f;

    if (nfull > 0) {
        // Issue one tile (8 async b32 ops: 4 for values, 4 for feat_ids).
        auto issue = [&](long long t, int buf) {
            const float* gv = values   + lo + t * TILE;
            const int*   gf = feat_ids + lo + t * TILE;
            #pragma unroll
            for (int j = 0; j < TILE / 32; ++j) {
                async_g2lds_b32(gv + j * 32 + lane, &s_vals[wave][buf][j * 32 + lane]);
                async_g2lds_b32(gf + j * 32 + lane, &s_feat[wave][buf][j * 32 + lane]);
            }
        };

        issue(0, 0);
        for (long long t = 0; t < nfull; ++t) {
            const int buf = (int)(t & 1);
            if (t + 1 < nfull) {
                // Reads of buf^1 (iteration t-1) must retire before overwrite.
                asm volatile("s_wait_dscnt 0x0" ::: "memory");
                issue(t + 1, buf ^ 1);
                // In-order async completion: <=8 outstanding => tile t arrived.
                asm volatile("s_wait_asynccnt 0x8" ::: "memory");
            } else {
                asm volatile("s_wait_asynccnt 0x0" ::: "memory");
            }
            const float* sv = s_vals[wave][buf];
            const int*   sf = s_feat[wave][buf];
            a0 += sv[lane     ] * w[sf[lane     ]];
            a1 += sv[lane + 32] * w[sf[lane + 32]];
            a2 += sv[lane + 64] * w[sf[lane + 64]];
            a3 += sv[lane + 96] * w[sf[lane + 96]];
        }
    }

    // Tail (n % 128) and short segments: direct coalesced strided loads.
    {
        long long i = lo + (nfull << 7) + lane;
        if (i < hi) {
            __builtin_prefetch(values   + i, 0, 0);   // gfx1250 global_prefetch_b8
            __builtin_prefetch(feat_ids + i, 0, 0);
        }
        for (; i < hi; i += 32) {
            a0 += values[i] * w[feat_ids[i]];
        }
    }

    float acc = (a0 + a1) + (a2 + a3);

    // Deterministic wave32 tree reduction (warpSize == 32 on gfx1250).
    #pragma unroll
    for (int m = 16; m >= 1; m >>= 1) {
        acc += __shfl_xor(acc, m, 32);
    }

    if (lane == 0) {
        out[row] = acc + b[0];
    }
}

extern "C" void kernel_launch(void* const* d_in, const int* in_sizes, int n_in,
                              void* d_out, int out_size, void* d_ws, size_t ws_size,
                              hipStream_t stream) {
    // Inputs (setup_inputs order): values, w, b, row_ids, feat_ids, batch
    const float* values   = (const float*)d_in[0];
    const float* w        = (const float*)d_in[1];
    const float* b        = (const float*)d_in[2];
    const int*   row_ids  = (const int*)d_in[3];
    const int*   feat_ids = (const int*)d_in[4];
    // batch (d_in[5]) is a device scalar; use out_size instead (== BATCH).

    float* out = (float*)d_out;
    const long long nnz = (long long)in_sizes[0];
    const int batch = out_size;

    // One wave (32 threads) per row; 256-thread blocks = 8 waves per block.
    const long long total_threads = (long long)batch * 32;
    const int block = 256;
    const int grid = (int)((total_threads + block - 1) / block);

    fm_linear_spmv_rowwave<<<grid, block, 0, stream>>>(
        values, w, b, row_ids, feat_ids, out, nnz, batch);
}